// Model_18932215841192
// MI455X (gfx1250) — compile-verified
//
#include <hip/hip_runtime.h>
#include <math.h>

// MI455X / gfx1250 fused S4D network:
//   grid = B (128) workgroups, block = 1024 threads = 32 wave32s.
//   wave = channel h (0..31), lane = SSM mode n (0..31).
//   Whole network runs in one persistent kernel; SSMs as exact recurrences
//   (states in VGPRs), 32x32 dense layer via V_WMMA_F32_16X16X4_F32 on
//   T=32 timestep micro-batches staged in LDS.
//
//   Mode-reduction strategy: instead of 5 shuffle(LDS-crossbar) ops per step,
//   each lane streams its per-step product into a wave-private LDS tile
//   (1 ds_store_b32 per step, not on the recurrence critical path); after the
//   chunk, lane tau reloads its row with 8 ds_load_b128 and reduces locally.
//   Wave-private tiles live in dynamic LDS (32 waves * 32*36 floats = 147 KB
//   of the 320 KB CDNA5 WGP pool).

typedef __attribute__((ext_vector_type(2))) float v2f;
typedef __attribute__((ext_vector_type(8))) float v8f;

#define HN 32      // model width
#define NN 32      // SSM state size
#define TT 32      // timestep chunk (micro-batch for the WMMA GEMM)
#define RS 36      // padded LDS row stride (floats): 16B-aligned, bank-safe
#define PS 36      // p-tile row stride (floats)

__device__ __forceinline__ float gelu_f(float v) {
    // exact GELU (matches jax.nn.gelu approximate=False)
    return 0.5f * v * (1.0f + erff(v * 0.7071067811865476f));
}

// Build per-(h,n) discrete SSM constants:
//   A   = -exp(log_A_real) + i*A_imag
//   a   = exp(dt*A)                      (recurrence multiplier)
//   coef= C*B*(exp(dt*A)-1)/A            (ZOH dB folded into C)
// Output: a_re,a_im and 2*coef_re, 2*coef_im (y uses 2*Re(coef*s)).
__device__ __forceinline__ void make_consts(
    const float* __restrict__ log_dt, const float* __restrict__ lAr,
    const float* __restrict__ Ai,     const float* __restrict__ Bre,
    const float* __restrict__ Bim,    const float* __restrict__ Cre,
    const float* __restrict__ Cim,
    int h, int n,
    float& ar, float& ai_, float& cr2, float& ci2)
{
    const int idx = h * NN + n;
    const float dt  = expf(log_dt[h]);
    const float Are = -expf(lAr[idx]);
    const float Aim = Ai[idx];
    const float dAr = dt * Are, dAi = dt * Aim;
    const float er  = expf(dAr);
    const float a_re = er * cosf(dAi);
    const float a_im = er * sinf(dAi);
    // t = C*B (complex)
    const float br = Bre[idx], bi = Bim[idx];
    const float cr = Cre[idx], ci = Cim[idx];
    const float t_re = cr * br - ci * bi;
    const float t_im = cr * bi + ci * br;
    // u = t * (a - 1)
    const float e_re = a_re - 1.0f, e_im = a_im;
    const float u_re = t_re * e_re - t_im * e_im;
    const float u_im = t_re * e_im + t_im * e_re;
    // coef = u / A = u * conj(A) / |A|^2
    const float inv = 1.0f / (Are * Are + Aim * Aim);
    const float co_re = (u_re * Are + u_im * Aim) * inv;
    const float co_im = (u_im * Are - u_re * Aim) * inv;
    ar = a_re; ai_ = a_im; cr2 = 2.0f * co_re; ci2 = 2.0f * co_im;
}

// One chunk of the diagonal-SSM scan for one wave (= one channel h).
// Lane tau holds input u[tau] in u_lanes (broadcast per step via readlane).
// Each lane owns one mode n's complex state. Per step: state update + one
// ds_store of p[tau][n] into the wave-private tile. After the loop, lane tau
// reduces its row p[tau][0..31] with 8x ds_load_b128 (no barrier needed:
// tile is wave-private and LDS ops are in-order within a wave).
template <int S>
__device__ __forceinline__ float ssm_scan(float u_lanes, float* pw,
                                          float ar, float ai, float cr2, float ci2,
                                          float Dh, float& sre, float& sim, int lane)
{
#pragma unroll
    for (int t = 0; t < S; ++t) {
        const float u  = __shfl(u_lanes, t, 32);     // readlane (VALU)
        const float nr = fmaf(ar, sre, fmaf(-ai, sim, u));
        const float ni = fmaf(ar, sim, ai * sre);
        sre = nr; sim = ni;
        pw[t * PS + lane] = fmaf(cr2, sre, -ci2 * sim);  // off critical path
    }
    float acc = 0.0f;
    const float4* row = (const float4*)&pw[lane * PS];   // lane tau's row
#pragma unroll
    for (int q = 0; q < 8; ++q) {
        const float4 p4 = row[q];
        acc += (p4.x + p4.y) + (p4.z + p4.w);
    }
    return fmaf(Dh, u_lanes, acc);                        // lane tau: + D*u[tau]
}

__device__ __forceinline__ float ssm_scan_dyn(float u_lanes, int steps, float* pw,
                                              float ar, float ai, float cr2, float ci2,
                                              float Dh, float& sre, float& sim, int lane)
{
    for (int t = 0; t < steps; ++t) {
        const float u  = __shfl(u_lanes, t, 32);
        const float nr = fmaf(ar, sre, fmaf(-ai, sim, u));
        const float ni = fmaf(ar, sim, ai * sre);
        sre = nr; sim = ni;
        pw[t * PS + lane] = fmaf(cr2, sre, -ci2 * sim);
    }
    float acc = 0.0f;
    const float4* row = (const float4*)&pw[lane * PS];
#pragma unroll
    for (int q = 0; q < 8; ++q) {
        const float4 p4 = row[q];
        acc += (p4.x + p4.y) + (p4.z + p4.w);
    }
    return fmaf(Dh, u_lanes, acc);   // lanes >= steps produce garbage; discarded
}

__global__ __launch_bounds__(1024, 1)
void s4d_fused_kernel(const float* __restrict__ x,
                      const float* __restrict__ W1, const float* __restrict__ b1,
                      const float* __restrict__ W2, const float* __restrict__ b2,
                      const float* __restrict__ W3, const float* __restrict__ b3,
                      const float* __restrict__ ldt1, const float* __restrict__ lAr1,
                      const float* __restrict__ Ai1,  const float* __restrict__ Br1,
                      const float* __restrict__ Bi1,  const float* __restrict__ Cr1,
                      const float* __restrict__ Ci1,  const float* __restrict__ D1,
                      const float* __restrict__ ldt2, const float* __restrict__ lAr2,
                      const float* __restrict__ Ai2,  const float* __restrict__ Br2,
                      const float* __restrict__ Bi2,  const float* __restrict__ Cr2,
                      const float* __restrict__ Ci2,  const float* __restrict__ D2,
                      float* __restrict__ out, int L)
{
    __shared__ __align__(16) float xs[TT];
    __shared__ __align__(16) float g1buf[TT * RS];  // gelu(SSM1 out)  [tau][h]
    __shared__ __align__(16) float h2buf[TT * RS];  // gelu(dense out) [tau][h]
    __shared__ __align__(16) float g2buf[TT * RS];  // gelu(SSM2 out)  [tau][h]
    __shared__ __align__(16) float w2s[HN * HN];    // W2 row-major [in][out]
    __shared__ __align__(16) float b2s[HN];
    __shared__ __align__(16) float w3s[HN];
    extern __shared__ __align__(16) float ptiles[]; // 32 waves * TT*PS floats

    const int tid  = threadIdx.x;
    const int lane = tid & 31;
    const int wid  = tid >> 5;
    const int h    = wid;                 // channel owned by this wave
    const int bidx = blockIdx.x;          // batch element

    float* pw = ptiles + wid * (TT * PS); // wave-private p tile

    if (tid < HN * HN) w2s[tid] = W2[tid];
    if (tid < HN) { b2s[tid] = b2[tid]; w3s[tid] = W3[tid]; }
    const float b3v = b3[0];

    const float w1h = W1[h], b1h = b1[h];
    const float D1h = D1[h], D2h = D2[h];

    float a1r, a1i, c1r, c1i, a2r, a2i, c2r, c2i;
    make_consts(ldt1, lAr1, Ai1, Br1, Bi1, Cr1, Ci1, h, lane, a1r, a1i, c1r, c1i);
    make_consts(ldt2, lAr2, Ai2, Br2, Bi2, Cr2, Ci2, h, lane, a2r, a2i, c2r, c2i);

    float s1r = 0.0f, s1i = 0.0f, s2r = 0.0f, s2i = 0.0f;

    const float* __restrict__ xb = x   + (size_t)bidx * L;
    float*       __restrict__ ob = out + (size_t)bidx * L;

    __syncthreads();

    for (int t0 = 0; t0 < L; t0 += TT) {
        const int steps = min(TT, L - t0);

        // ---- front-end: xl = 20*log10(max(|x|,1e-4)) staged in LDS ----
        if (wid == 0 && lane < steps) {
            const float xv = xb[t0 + lane];
            xs[lane] = 20.0f * log10f(fmaxf(fabsf(xv), 1e-4f));
        }
        __syncthreads();

        // ---- lift 1->H + GELU (per wave: its own channel) ----
        float u1 = 0.0f;
        if (lane < steps) u1 = gelu_f(fmaf(xs[lane], w1h, b1h));

        // ---- SSM layer 1 (scan over chunk) ----
        const float y1 = (steps == TT)
            ? ssm_scan<TT>(u1, pw, a1r, a1i, c1r, c1i, D1h, s1r, s1i, lane)
            : ssm_scan_dyn(u1, steps, pw, a1r, a1i, c1r, c1i, D1h, s1r, s1i, lane);
        if (lane < steps) g1buf[lane * RS + h] = gelu_f(y1);
        __syncthreads();

        // ---- 32x32 dense layer via f32 WMMA: [T x H] @ [H x H] ----
        // 4 output tiles of 16x16, one wave each; K=32 as 8 slices of K=4.
        if (wid < 4) {
            const int ti = wid >> 1;          // tau-tile
            const int tj = wid & 1;           // out-channel tile
            const int m  = lane & 15;
            const int hh = lane >> 4;         // lane half selects K sub-pair
            v8f c = {0.f, 0.f, 0.f, 0.f, 0.f, 0.f, 0.f, 0.f};
#pragma unroll
            for (int s = 0; s < 8; ++s) {
                const int k0  = 4 * s + 2 * hh;
                const int row = 16 * ti + m;  // tau
                v2f av, bv;
                av.x = g1buf[row * RS + k0];
                av.y = g1buf[row * RS + k0 + 1];
                bv.x = w2s[(k0)     * HN + 16 * tj + m];
                bv.y = w2s[(k0 + 1) * HN + 16 * tj + m];
                c = __builtin_amdgcn_wmma_f32_16x16x4_f32(
                        false, av, false, bv, (short)0, c, false, false);
            }
            const float bj = b2s[16 * tj + m];
#pragma unroll
            for (int v = 0; v < 8; ++v) {
                const int tau = 16 * ti + v + 8 * hh;   // ISA C/D layout
                if (tau < steps)
                    h2buf[tau * RS + 16 * tj + m] = gelu_f(c[v] + bj);
            }
        }
        __syncthreads();

        // ---- SSM layer 2 ----
        float u2 = 0.0f;
        if (lane < steps) u2 = h2buf[lane * RS + h];
        const float y2 = (steps == TT)
            ? ssm_scan<TT>(u2, pw, a2r, a2i, c2r, c2i, D2h, s2r, s2i, lane)
            : ssm_scan_dyn(u2, steps, pw, a2r, a2i, c2r, c2i, D2h, s2r, s2i, lane);
        if (lane < steps) g2buf[lane * RS + h] = gelu_f(y2);
        __syncthreads();

        // ---- head: y = g2 . W3 + b3 ; out = 10^(y/20) ----
        if (wid == 0 && lane < steps) {
            const float4* gr = (const float4*)&g2buf[lane * RS]; // RS*4=144B, 16B aligned
            const float4* wr = (const float4*)w3s;
            float acc = 0.0f;
#pragma unroll
            for (int q = 0; q < 8; ++q) {
                const float4 g = gr[q];
                const float4 w = wr[q];
                acc += g.x * w.x + g.y * w.y + g.z * w.z + g.w * w.w;
            }
            // 10^(y/20) = 2^(y * log2(10)/20)
            ob[t0 + lane] = exp2f((acc + b3v) * 0.16609640474436813f);
        }
        // next chunk's xs/g1/h2/g2 writes are fenced by the barriers above
    }
}

extern "C" void kernel_launch(void* const* d_in, const int* in_sizes, int n_in,
                              void* d_out, int out_size, void* d_ws, size_t ws_size,
                              hipStream_t stream)
{
    (void)n_in; (void)d_ws; (void)ws_size; (void)out_size;
    const float* x   = (const float*)d_in[0];
    const float* W1  = (const float*)d_in[1];
    const float* b1  = (const float*)d_in[2];
    const float* W2  = (const float*)d_in[3];
    const float* b2  = (const float*)d_in[4];
    const float* W3  = (const float*)d_in[5];
    const float* b3  = (const float*)d_in[6];
    const float* ldt1 = (const float*)d_in[7];
    const float* lAr1 = (const float*)d_in[8];
    const float* Ai1  = (const float*)d_in[9];
    const float* Br1  = (const float*)d_in[10];
    const float* Bi1  = (const float*)d_in[11];
    const float* Cr1  = (const float*)d_in[12];
    const float* Ci1  = (const float*)d_in[13];
    const float* D1   = (const float*)d_in[14];
    const float* ldt2 = (const float*)d_in[15];
    const float* lAr2 = (const float*)d_in[16];
    const float* Ai2  = (const float*)d_in[17];
    const float* Br2  = (const float*)d_in[18];
    const float* Bi2  = (const float*)d_in[19];
    const float* Cr2  = (const float*)d_in[20];
    const float* Ci2  = (const float*)d_in[21];
    const float* D2   = (const float*)d_in[22];
    float* out = (float*)d_out;

    const int B = 128;                 // batch (reference fixes B=128)
    const int L = in_sizes[0] / B;     // sequence length

    const size_t dyn_lds = (size_t)32 * TT * PS * sizeof(float); // 147456 B p-tiles

    hipLaunchKernelGGL(s4d_fused_kernel, dim3(B), dim3(1024), dyn_lds, stream,
                       x, W1, b1, W2, b2, W3, b3,
                       ldt1, lAr1, Ai1, Br1, Bi1, Cr1, Ci1, D1,
                       ldt2, lAr2, Ai2, Br2, Bi2, Cr2, Ci2, D2,
                       out, L);
}